// DCTFrequencySplitter_28346784153711
// MI455X (gfx1250) — compile-verified
//
#include <hip/hip_runtime.h>
#include <math.h>

#define HH 14
#define WW 14
#define NP 196          // patches per image
#define NTOK 197
#define DD 768
#define BATCH 128
#define TSTRIDE 200     // padded row stride of T (floats); even -> 8B alignment for even k
#define TSLOT 41600     // 208*200 floats reserved per T matrix in workspace
#define LDS_FLOATS (208 * TSTRIDE)   // padded so tile-12 garbage rows stay in-bounds

typedef __attribute__((ext_vector_type(2))) float v2f;
typedef __attribute__((ext_vector_type(8))) float v8f;

// ---------------------------------------------------------------------------
// Kernel 1: build the 196x196 circulant operators T_low / T_high from the
// mask parameters.  T[p,q] = (1/196) * sum_k m[k] * cos(2*pi*k.(p-q)/14)
// = K[(p-q) mod 14x14] with K = Re(ifft2(m)).
// ---------------------------------------------------------------------------
__global__ void prep_masks_kernel(const float* __restrict__ lp,
                                  const float* __restrict__ hp,
                                  float* __restrict__ Tlo,
                                  float* __restrict__ Thi) {
    __shared__ float ml[NP], mh[NP], Kl[NP], Kh[NP];
    const int tid = threadIdx.x;
    if (tid < NP) {
        const float y = (float)(tid / WW);
        const float x = (float)(tid % WW);
        {   // low mask
            float ch = lp[0], cw = lp[1];
            float r = fminf(fmaxf(lp[2], 1.0f), 7.0f);
            float s = fminf(fmaxf(lp[3], 0.5f), 10.0f);
            float d2 = (y - ch) * (y - ch) + (x - cw) * (x - cw);
            float dist = sqrtf(d2 + 1e-12f);
            ml[tid] = expf(-powf(dist / r, s));
        }
        {   // high mask
            float ch = hp[0], cw = hp[1];
            float r = fminf(fmaxf(hp[2], 1.0f), 7.0f);
            float s = fminf(fmaxf(hp[3], 0.5f), 10.0f);
            float d2 = (y - ch) * (y - ch) + (x - cw) * (x - cw);
            float dist = sqrtf(d2 + 1e-12f);
            mh[tid] = 1.0f - expf(-powf(dist / r, s));
        }
    }
    __syncthreads();
    if (tid < NP) {
        const int ry = tid / WW, rx = tid % WW;
        float sl = 0.0f, sh = 0.0f;
        for (int ky = 0; ky < HH; ++ky) {
            for (int kx = 0; kx < WW; ++kx) {
                float ang = (2.0f * 3.14159265358979323846f / 14.0f) *
                            (float)(ky * ry + kx * rx);
                float c = cosf(ang);
                int k = ky * WW + kx;
                sl += ml[k] * c;
                sh += mh[k] * c;
            }
        }
        Kl[tid] = sl * (1.0f / (float)NP);
        Kh[tid] = sh * (1.0f / (float)NP);
    }
    __syncthreads();
    for (int idx = tid; idx < NP * NP; idx += blockDim.x) {
        int p = idx / NP, q = idx % NP;
        int ry = ((p / WW) - (q / WW) + HH) % HH;
        int rx = ((p % WW) - (q % WW) + WW) % WW;
        Tlo[p * TSTRIDE + q] = Kl[ry * WW + rx];
        Thi[p * TSTRIDE + q] = Kh[ry * WW + rx];
    }
}

// ---------------------------------------------------------------------------
// Kernel 2: fused mean-over-patches + 768->192 ReLU + 192->2 sigmoid gate.
// scales[0*128+b] = sigmoid(gate0)*sigmoid(alpha_low)
// scales[1*128+b] = sigmoid(gate1)*sigmoid(alpha_high)
// ---------------------------------------------------------------------------
__global__ void gate_kernel(const float* __restrict__ x,
                            const float* __restrict__ w1,
                            const float* __restrict__ b1,
                            const float* __restrict__ w2,
                            const float* __restrict__ b2,
                            const float* __restrict__ alo,
                            const float* __restrict__ ahi,
                            float* __restrict__ scales) {
    __shared__ float g[DD];
    __shared__ float h[192];
    const int b = blockIdx.x;
    const int tid = threadIdx.x;
    const float* xb = x + (size_t)b * NTOK * DD + DD;   // skip cls row
    for (int c = tid; c < DD; c += blockDim.x) {
        float s = 0.0f;
        for (int n = 0; n < NP; ++n) s += xb[(size_t)n * DD + c];
        g[c] = s * (1.0f / (float)NP);
    }
    __syncthreads();
    if (tid < 192) {
        float s = b1[tid];
        for (int c = 0; c < DD; ++c) s += g[c] * w1[c * 192 + tid];
        h[tid] = fmaxf(s, 0.0f);
    }
    __syncthreads();
    if (tid < 2) {
        float s = b2[tid];
        for (int j = 0; j < 192; ++j) s += h[j] * w2[j * 2 + tid];
        float gate = 1.0f / (1.0f + expf(-s));
        float a = (tid == 0) ? alo[0] : ahi[0];
        float av = 1.0f / (1.0f + expf(-a));
        scales[tid * BATCH + b] = gate * av;
    }
}

// ---------------------------------------------------------------------------
// Kernel 3: copy cls token row (unchanged) into both outputs.
// ---------------------------------------------------------------------------
__global__ void cls_copy_kernel(const float* __restrict__ x,
                                float* __restrict__ out) {
    const int b = blockIdx.x;
    const float* src = x + (size_t)b * NTOK * DD;
    float* o0 = out + (size_t)b * NTOK * DD;
    float* o1 = out + (size_t)BATCH * NTOK * DD + (size_t)b * NTOK * DD;
    for (int d = threadIdx.x; d < DD; d += blockDim.x) {
        float v = src[d];
        o0[d] = v;
        o1[d] = v;
    }
}

// ---------------------------------------------------------------------------
// WMMA GEMM core for one wave: TN M-tiles of 16 rows starting at mbase,
// one 16-column tile at `col`.  K = 196 in 49 steps of 4 via
// V_WMMA_F32_16X16X4_F32.  Ts is the LDS-resident T matrix (row-major,
// stride TSTRIDE floats).
// ---------------------------------------------------------------------------
template <int TN>
__device__ __forceinline__ void gemm_core(const float* __restrict__ Ts,
                                          const float* __restrict__ xb,
                                          float* __restrict__ ob,
                                          float s, int mbase, int col,
                                          int lrow, int khalf, int rhi) {
    v8f acc[TN];
#pragma unroll
    for (int t = 0; t < TN; ++t) acc[t] = (v8f){0, 0, 0, 0, 0, 0, 0, 0};

    // Per-lane A base: row = mbase + lrow, K offset khalf (0 or 2).
    const float* arow = Ts + (size_t)(mbase + lrow) * TSTRIDE + khalf;

    for (int k = 0; k < NP; k += 4) {              // 49 K-steps
        const int kk = k + khalf;
        // B fragment (4x16): lanes 0-15 rows k/k+1, lanes 16-31 rows k+2/k+3
        v2f bf;
        bf.x = xb[(size_t)kk * DD + col];
        bf.y = xb[(size_t)(kk + 1) * DD + col];
#pragma unroll
        for (int t = 0; t < TN; ++t) {
            // A fragment (16x4): two consecutive floats, 8B-aligned ds_load_b64
            const v2f af = *(const v2f*)__builtin_assume_aligned(
                arow + (size_t)t * 16 * TSTRIDE + k, 8);
            acc[t] = __builtin_amdgcn_wmma_f32_16x16x4_f32(
                false, af, false, bf, (short)0, acc[t], false, false);
        }
    }

    // C/D layout: VGPR v holds M = mbase + t*16 + v + (lane>=16 ? 8 : 0)
#pragma unroll
    for (int t = 0; t < TN; ++t) {
#pragma unroll
        for (int v = 0; v < 8; ++v) {
            const int row = mbase + t * 16 + v + rhi;
            if (row < NP) ob[(size_t)row * DD + col] = acc[t][v] * s;
        }
    }
}

// ---------------------------------------------------------------------------
// Kernel 4: batched fp32 WMMA GEMM.  blockIdx = (batch, mask).
// Out[b] (196x768) = scale * T (196x196) @ X[b] (196x768)
// 8 waves = 4 column-groups x 2 M-groups (tiles 0-6 and 7-12); each strip
// covers 64 columns, 12 strips over 768.
// ---------------------------------------------------------------------------
__global__ void __launch_bounds__(256)
freq_gemm_kernel(const float* __restrict__ x,
                 const float* __restrict__ Tmats,
                 const float* __restrict__ scales,
                 float* __restrict__ out) {
    extern __shared__ float Ts[];                  // 208 x TSTRIDE floats
    const int bx = blockIdx.x;
    const int b = bx >> 1;
    const int mask = bx & 1;
    const int tid = threadIdx.x;

    // Stage this mask's T matrix into LDS (rows 0..195 valid).
    const float* Tsrc = Tmats + (size_t)mask * TSLOT;
    for (int i = tid; i < NP * TSTRIDE; i += 256) Ts[i] = Tsrc[i];
    __syncthreads();

    const int wave = tid >> 5;
    const int lane = tid & 31;
    const int colgroup = wave >> 1;                // 0..3
    const int mgroup = wave & 1;                   // 0: tiles 0-6, 1: tiles 7-12
    const int lrow = lane & 15;
    const int khalf = (lane >> 4) << 1;            // lanes 16-31 take K+2
    const int rhi = (lane >> 4) << 3;              // C/D: lanes 16-31 hold M+8

    const float s = scales[mask * BATCH + b];
    const float* xb = x + (size_t)b * NTOK * DD + DD;                  // patches
    float* ob = out + (size_t)mask * BATCH * NTOK * DD +
                (size_t)b * NTOK * DD + DD;                            // patch rows

    for (int strip = 0; strip < 12; ++strip) {     // 12 * 64 = 768 cols
        const int col = strip * 64 + colgroup * 16 + lrow;
        if (mgroup == 0) {
            gemm_core<7>(Ts, xb, ob, s, 0, col, lrow, khalf, rhi);
        } else {
            gemm_core<6>(Ts, xb, ob, s, 112, col, lrow, khalf, rhi);
        }
    }
}

// ---------------------------------------------------------------------------
// Launch
// ---------------------------------------------------------------------------
extern "C" void kernel_launch(void* const* d_in, const int* in_sizes, int n_in,
                              void* d_out, int out_size, void* d_ws, size_t ws_size,
                              hipStream_t stream) {
    const float* x   = (const float*)d_in[0];
    const float* lp  = (const float*)d_in[1];
    const float* hp  = (const float*)d_in[2];
    const float* alo = (const float*)d_in[3];
    const float* ahi = (const float*)d_in[4];
    const float* w1  = (const float*)d_in[5];
    const float* b1  = (const float*)d_in[6];
    const float* w2  = (const float*)d_in[7];
    const float* b2  = (const float*)d_in[8];
    float* out = (float*)d_out;

    float* ws     = (float*)d_ws;
    float* Tlo    = ws;                // TSLOT floats
    float* Thi    = ws + TSLOT;        // TSLOT floats
    float* scales = ws + 2 * TSLOT;    // 256 floats

    prep_masks_kernel<<<1, 256, 0, stream>>>(lp, hp, Tlo, Thi);
    gate_kernel<<<BATCH, 256, 0, stream>>>(x, w1, b1, w2, b2, alo, ahi, scales);
    cls_copy_kernel<<<BATCH, 256, 0, stream>>>(x, out);
    freq_gemm_kernel<<<BATCH * 2, 256, LDS_FLOATS * sizeof(float), stream>>>(
        x, ws, scales, out);
}